// Midi_loss_7310034338519
// MI455X (gfx1250) — compile-verified
//
#include <hip/hip_runtime.h>
#include <hip/hip_bf16.h>

// MIDI contour loss, fused single-pass kernel for gfx1250 (MI455X).
//
// Problem sizes: B=32 samples, L=4096 frames, N=128 notes, SEG=32 frames/slot.
// Total input = 2.25 MB -> latency bound at 23.3 TB/s HBM; structure of the
// computation (interval masks) lets us avoid materializing the (N,B,L) mask
// entirely. One workgroup per sample (128 threads = 4 wave32), thread k owns
// note-slot k. Final per-block reduction of the 128 per-note losses is done
// on the matrix engine: V_WMMA_F32_16X16X4_F32 with an all-ones A matrix
// yields column sums of a 4x16 f32 tile with f32 accumulation.

typedef __attribute__((ext_vector_type(2))) float v2f;
typedef __attribute__((ext_vector_type(8))) float v8f;

#define B_N 32
#define L_N 4096
#define N_N 128
#define SEG_N (L_N / N_N)   // 32 frames per note slot

__global__ void midi_loss_zero_kernel(float* __restrict__ out) {
    out[0] = 0.0f;
    out[1] = 0.0f;
}

__global__ __launch_bounds__(N_N) void midi_loss_kernel(
    const float* __restrict__ gen_f0, const float* __restrict__ t_f0,
    const float* __restrict__ gen_lo, const float* __restrict__ t_lo,
    const unsigned char* __restrict__ onsets,
    const unsigned char* __restrict__ offsets,
    float* __restrict__ out)
{
    __shared__ float pc[N_N];   // per-note pitch loss contributions
    __shared__ float lc[N_N];   // per-note loudness loss contributions

    const int b = blockIdx.x;   // sample
    const int k = threadIdx.x;  // note slot

    const float* gf = gen_f0 + b * L_N;
    const float* tf = t_f0 + b * L_N;
    const float* gl = gen_lo + b * L_N;
    const float* tl = t_lo + b * L_N;
    const unsigned char* onb = onsets + b * L_N;
    const unsigned char* offb = offsets + b * L_N;

    const int base = k * SEG_N;

    // Pull this thread's slot toward the caches early (global_prefetch_b8).
    __builtin_prefetch(gf + base, 0, 1);
    __builtin_prefetch(tf + base, 0, 1);
    __builtin_prefetch(gl + base, 0, 1);
    __builtin_prefetch(tl + base, 0, 1);

    // First True position in this slot for onset / offset. Backward scan so
    // the surviving write is the smallest index (branch-free after unroll).
    int onp = base + SEG_N;
    int offp = base + SEG_N;
#pragma unroll
    for (int t = SEG_N - 1; t >= 0; --t) {
        if (onb[base + t]) onp = base + t;
        if (offb[base + t]) offp = base + t;
    }

    // Interval sums over [onp, offp) for the four signals (<= 32 elements).
    float sgf = 0.0f, stf = 0.0f, sgl = 0.0f, stl = 0.0f;
    for (int t = onp; t < offp; ++t) {
        sgf += gf[t];
        stf += tf[t];
        sgl += gl[t];
        stl += tl[t];
    }

    // mean_gen - mean_tgt = (Sg - St) / (cnt + 1e-6 * L)
    const float cnt = (float)(offp - onp);
    const float denom = cnt + 1e-6f * (float)L_N;
    const float dp = fabsf((sgf - stf) / denom) - 0.5f;  // F0_THRESHOLD
    const float dl = fabsf((sgl - stl) / denom) - 0.5f;  // LO_THRESHOLD
    pc[k] = dp > 0.0f ? dp : 0.0f;
    lc[k] = dl > 0.0f ? dl : 0.0f;

    __syncthreads();

    // Wave 0 reduces the 128 values per array with the matrix engine.
    // A = ones(16x4), B = 4x16 tile of data => D[i][j] = sum_k B[k][j] (+C).
    // Every D row equals the column sums, so summing D's VGPR0 across all 32
    // lanes gives 2x the tile total (lanes 0-15 = row M=0, lanes 16-31 = M=8).
    if (threadIdx.x < 32) {
        const int lane = threadIdx.x;
        v2f a;
        a.x = 1.0f;
        a.y = 1.0f;
        v8f accP = {};
        v8f accL = {};
#pragma unroll
        for (int c = 0; c < 2; ++c) {
            v2f bp, bl;
            bp.x = pc[c * 64 + lane];
            bp.y = pc[c * 64 + 32 + lane];
            bl.x = lc[c * 64 + lane];
            bl.y = lc[c * 64 + 32 + lane];
            accP = __builtin_amdgcn_wmma_f32_16x16x4_f32(
                false, a, false, bp, (short)0, accP, false, false);
            accL = __builtin_amdgcn_wmma_f32_16x16x4_f32(
                false, a, false, bl, (short)0, accL, false, false);
        }

        float sp = accP[0];
        float sl = accL[0];
#pragma unroll
        for (int w = 16; w >= 1; w >>= 1) {
            sp += __shfl_xor(sp, w, 32);
            sl += __shfl_xor(sl, w, 32);
        }

        if (lane == 0) {
            // /2 for the duplicated-rows double count, /(N*B) for the mean.
            const float scale = 0.5f / (float)(N_N * B_N);
            atomicAdd(&out[0], sp * scale);
            atomicAdd(&out[1], sl * scale);
        }
    }
}

extern "C" void kernel_launch(void* const* d_in, const int* in_sizes, int n_in,
                              void* d_out, int out_size, void* d_ws, size_t ws_size,
                              hipStream_t stream) {
    (void)in_sizes; (void)n_in; (void)out_size; (void)d_ws; (void)ws_size;

    const float* gen_f0 = (const float*)d_in[0];
    const float* t_f0   = (const float*)d_in[1];
    const float* gen_lo = (const float*)d_in[2];
    const float* t_lo   = (const float*)d_in[3];
    const unsigned char* onsets  = (const unsigned char*)d_in[4];
    const unsigned char* offsets = (const unsigned char*)d_in[5];
    float* out = (float*)d_out;

    midi_loss_zero_kernel<<<1, 1, 0, stream>>>(out);
    midi_loss_kernel<<<B_N, N_N, 0, stream>>>(gen_f0, t_f0, gen_lo, t_lo,
                                              onsets, offsets, out);
}